// Transfer_55070070670135
// MI455X (gfx1250) — compile-verified
//
#include <hip/hip_runtime.h>
#include <math.h>

typedef __attribute__((ext_vector_type(2))) float v2f;
typedef __attribute__((ext_vector_type(4))) float v4f;
typedef __attribute__((ext_vector_type(8))) float v8f;

#define EPS_F 1e-8f
#define NEG_BIG (-3.402823466e38f)

static constexpr int Bn   = 1024;   // batch
static constexpr int Sn   = 512;    // sequence
static constexpr int Fn   = 256;    // features
static constexpr int TWOF = 512;    // 2*F
static constexpr int Mn   = 4096;   // memory rows

// ---------------------------------------------------------------------------
// Kernel 1: per-(b,f) concept stats in ONE pass over x, float4-vectorized.
//   c_mean[b,f] = sum_s (ce[s,f]/S) * x[b,s,f]
//   c_std [b,f] = sqrt( mean_s x^2 - 2*c_mean*mean_s x + c_mean^2 + EPS )
// 1024 blocks (b) x 64 threads; each thread owns 4 consecutive features,
// so every lane issues one global_load_b128 per row (512 B / wave / instr).
// ---------------------------------------------------------------------------
__global__ void stats_kernel(const float* __restrict__ x,
                             const float* __restrict__ ce,
                             float* __restrict__ cpt) {
    const int b = blockIdx.x;
    const int t = threadIdx.x;               // 0..63 -> float4 column
    const float* xb = x  + (size_t)b * Sn * Fn + t * 4;
    const float* cb = ce + t * 4;
    v4f sw = {}, sx = {}, sx2 = {};
#pragma unroll 4
    for (int s = 0; s < Sn; ++s) {
        v4f xv = *(const v4f*)(xb + (size_t)s * Fn);
        v4f w  = *(const v4f*)(cb + (size_t)s * Fn);
        sw  += w * xv;
        sx  += xv;
        sx2 += xv * xv;
    }
    const float invS = 1.0f / (float)Sn;
    v4f cmean = sw * invS;
    v4f var = sx2 * invS - 2.0f * cmean * (sx * invS) + cmean * cmean;
    v4f cstd;
    cstd.x = sqrtf(var.x + EPS_F);
    cstd.y = sqrtf(var.y + EPS_F);
    cstd.z = sqrtf(var.z + EPS_F);
    cstd.w = sqrtf(var.w + EPS_F);
    *(v4f*)(cpt + (size_t)b * TWOF + t * 4)      = cmean;
    *(v4f*)(cpt + (size_t)b * TWOF + Fn + t * 4) = cstd;
}

// ---------------------------------------------------------------------------
// Kernel 2: mn[m] = ||memory[m]||^2.  4096 blocks x 128 threads (float4 each).
// ---------------------------------------------------------------------------
__global__ void mn_kernel(const float* __restrict__ mem,
                          float* __restrict__ mn) {
    const int m = blockIdx.x;
    const int t = threadIdx.x;                      // 0..127
    v4f v = *(const v4f*)(mem + (size_t)m * TWOF + t * 4);
    float s = v.x * v.x + v.y * v.y + v.z * v.z + v.w * v.w;
#pragma unroll
    for (int off = 16; off >= 1; off >>= 1) s += __shfl_xor(s, off, 32);
    __shared__ float red[4];
    if ((t & 31) == 0) red[t >> 5] = s;
    __syncthreads();
    if (t == 0) mn[m] = red[0] + red[1] + red[2] + red[3];
}

// ---------------------------------------------------------------------------
// Kernel 3: sim = 2*(concept @ memory^T) - mn  via V_WMMA_F32_16X16X4_F32,
// fused with per-row argmax.  (-||c||^2 is row-constant: argmax-invariant.)
// grid = B/16 blocks; block = 8 waves; wave handles 16x16 tiles, striding
// column tiles by 8.  K loop: 512/4 = 128 WMMA ops per tile.
// A layout (16x4 f32): lane L holds A[L%16][k0 + 2*(L/16) + {0,1}]
// B layout (4x16 f32): lane L holds B[k0 + 2*(L/16) + {0,1}][L%16]
//                      = memory[m0 + L%16][k0 + 2*(L/16) + {0,1}]
// D layout: vgpr r of lane L = sim[row r + 8*(L/16)][col L%16]
// ---------------------------------------------------------------------------
__global__ void argmax_kernel(const float* __restrict__ cpt,
                              const float* __restrict__ mem,
                              const float* __restrict__ mn,
                              int* __restrict__ indices) {
    const int lane = threadIdx.x & 31;
    const int wave = threadIdx.x >> 5;      // 0..7
    const int half = lane >> 4;             // 0/1
    const int l16  = lane & 15;
    const int b0   = blockIdx.x * 16;

    const float* arow = cpt + (size_t)(b0 + l16) * TWOF + 2 * half;

    float bestv[8];
    int   besti[8];
#pragma unroll
    for (int r = 0; r < 8; ++r) { bestv[r] = NEG_BIG; besti[r] = 0; }

    for (int mt = wave; mt < Mn / 16; mt += 8) {
        const int m0 = mt * 16;
        const float* brow = mem + (size_t)(m0 + l16) * TWOF + 2 * half;
        v8f acc = {};
#pragma unroll 4
        for (int k0 = 0; k0 < TWOF; k0 += 4) {
            v2f a  = *(const v2f*)(arow + k0);
            v2f bb = *(const v2f*)(brow + k0);
            acc = __builtin_amdgcn_wmma_f32_16x16x4_f32(
                false, a, false, bb, (short)0, acc, false, false);
        }
        const float mnv = mn[m0 + l16];
        const int   col = m0 + l16;
#pragma unroll
        for (int r = 0; r < 8; ++r) {
            float v = fmaf(2.0f, acc[r], -mnv);
            if (v > bestv[r]) { bestv[r] = v; besti[r] = col; }
        }
    }

    // argmax across the 16 lanes of each half-wave (columns), per row r
#pragma unroll
    for (int r = 0; r < 8; ++r) {
        float v = bestv[r]; int i = besti[r];
#pragma unroll
        for (int off = 8; off >= 1; off >>= 1) {
            float ov = __shfl_xor(v, off, 32);
            int   oi = __shfl_xor(i, off, 32);
            if (ov > v || (ov == v && oi < i)) { v = ov; i = oi; }
        }
        bestv[r] = v; besti[r] = i;
    }

    __shared__ float sv[8][16];
    __shared__ int   si[8][16];
    if (l16 == 0) {
#pragma unroll
        for (int r = 0; r < 8; ++r) {
            sv[wave][r + 8 * half] = bestv[r];
            si[wave][r + 8 * half] = besti[r];
        }
    }
    __syncthreads();
    if (threadIdx.x < 16) {
        float v = sv[0][threadIdx.x]; int i = si[0][threadIdx.x];
#pragma unroll
        for (int w = 1; w < 8; ++w) {
            float ov = sv[w][threadIdx.x]; int oi = si[w][threadIdx.x];
            if (ov > v || (ov == v && oi < i)) { v = ov; i = oi; }
        }
        indices[b0 + threadIdx.x] = i;
    }
}

// ---------------------------------------------------------------------------
// Kernel 4: precompute scale/bias [B,F] from selected memory rows.
// ---------------------------------------------------------------------------
__global__ void scalebias_kernel(const float* __restrict__ cpt,
                                 const float* __restrict__ mem,
                                 const int* __restrict__ indices,
                                 float* __restrict__ scale,
                                 float* __restrict__ bias) {
    int e = blockIdx.x * blockDim.x + threadIdx.x;
    if (e >= Bn * Fn) return;
    int b = e >> 8;          // / Fn
    int f = e & (Fn - 1);
    int id = indices[b];
    float cm = cpt[(size_t)b * TWOF + f];
    float cs = cpt[(size_t)b * TWOF + Fn + f];
    float ms = mem[(size_t)id * TWOF + Fn + f];
    float mb = mem[(size_t)id * TWOF + f];
    float sc = ms / cs;
    scale[e] = sc;
    bias[e]  = fmaf(-cm, sc, mb);
}

// ---------------------------------------------------------------------------
// Kernel 5: out = x*scale + bias, bandwidth-critical stream (float4 fma).
// grid = B*8 blocks; block = 64 threads (one float4 column each); scale/bias
// held in registers across 64 S-rows.
// ---------------------------------------------------------------------------
__global__ void apply_kernel(const float* __restrict__ x,
                             const float* __restrict__ scale,
                             const float* __restrict__ bias,
                             float* __restrict__ out) {
    const int blk = blockIdx.x;
    const int b   = blk >> 3;
    const int sc  = blk & 7;
    const int t   = threadIdx.x;             // 0..63 -> float4 column
    const int SPC = Sn / 8;                  // 64 rows per block

    v4f scl = *(const v4f*)(scale + (size_t)b * Fn + t * 4);
    v4f bs  = *(const v4f*)(bias  + (size_t)b * Fn + t * 4);
    size_t base = ((size_t)b * Sn + (size_t)sc * SPC) * Fn + t * 4;
#pragma unroll 4
    for (int s = 0; s < SPC; ++s) {
        v4f xv = *(const v4f*)(x + base);
        v4f o;
        o.x = fmaf(xv.x, scl.x, bs.x);
        o.y = fmaf(xv.y, scl.y, bs.y);
        o.z = fmaf(xv.z, scl.z, bs.z);
        o.w = fmaf(xv.w, scl.w, bs.w);
        *(v4f*)(out + base) = o;
        base += Fn;
    }
}

// ---------------------------------------------------------------------------
extern "C" void kernel_launch(void* const* d_in, const int* in_sizes, int n_in,
                              void* d_out, int out_size, void* d_ws, size_t ws_size,
                              hipStream_t stream) {
    const float* x   = (const float*)d_in[0];   // [B,S,F]
    const float* mem = (const float*)d_in[1];   // [M,2F]
    const float* ce  = (const float*)d_in[2];   // [S,F]
    float* out = (float*)d_out;

    char* ws = (char*)d_ws;
    float* cpt     = (float*)ws;                                   // B*2F floats
    float* mn      = (float*)(ws + (size_t)Bn * TWOF * 4);         // M floats
    int*   indices = (int*)  (ws + ((size_t)Bn * TWOF + Mn) * 4);  // B ints
    float* scale   = (float*)(ws + ((size_t)Bn * TWOF + Mn + Bn) * 4);       // B*F
    float* bias    = scale + (size_t)Bn * Fn;                                // B*F

    stats_kernel    <<<Bn,            64,  0, stream>>>(x, ce, cpt);
    mn_kernel       <<<Mn,            128, 0, stream>>>(mem, mn);
    argmax_kernel   <<<Bn / 16,       256, 0, stream>>>(cpt, mem, mn, indices);
    scalebias_kernel<<<(Bn * Fn + 255) / 256, 256, 0, stream>>>(cpt, mem, indices, scale, bias);
    apply_kernel    <<<Bn * 8,        64,  0, stream>>>(x, scale, bias, out);
}